// Forward_Model_27616639714067
// MI455X (gfx1250) — compile-verified
//
#include <hip/hip_runtime.h>
#include <stdint.h>

typedef __attribute__((ext_vector_type(2))) float v2f;
typedef __attribute__((ext_vector_type(8))) float v8f;

#define IMG   64
#define CH    4096
#define PW    72                 // padded LDS row stride in floats (288B, 16B aligned)
#define PH    66
#define TILE_FLOATS (PH * PW)    // 4752 floats = 19008 B

__device__ __forceinline__ uint32_t lds_addr_of(const void* p) {
  // generic shared pointer: low 32 bits == LDS offset (aperture in high bits)
  return (uint32_t)(uintptr_t)p;
}

// Stage one 64x64 f32 channel image into the padded LDS tile with the async
// global->LDS path (ASYNCcnt), zeroing the 1-element conv halo.
__device__ __forceinline__ void fill_tile_async(float* tile, const float* gsrc) {
  const int tid = threadIdx.x;
  // zero borders: row 0, row 65, col 3, col 68 (interior starts at col 4)
  if (tid < 260) {
    int pr, pc;
    if (tid < 66)       { pr = 0;              pc = 3 + tid; }
    else if (tid < 132) { pr = PH - 1;         pc = 3 + (tid - 66); }
    else if (tid < 196) { pr = 1 + (tid - 132); pc = 3; }
    else                { pr = 1 + (tid - 196); pc = 68; }
    tile[pr * PW + pc] = 0.0f;
  }
  const uint32_t tbase = lds_addr_of(tile);
#pragma unroll
  for (int k = 0; k < 4; ++k) {
    int i   = tid + k * 256;          // 1024 chunks of 16B
    int row = i >> 4;                 // image row
    int col = (i & 15) << 2;          // image col (floats)
    uint32_t lds_b = tbase + (uint32_t)(((row + 1) * PW + 4 + col) * 4);
    uint32_t voff  = (uint32_t)(i << 4);
    asm volatile("global_load_async_to_lds_b128 %0, %1, %2"
                 :: "v"(lds_b), "v"(voff), "s"(gsrc) : "memory");
  }
  asm volatile("s_wait_asynccnt 0" ::: "memory");
  __syncthreads();
}

// clamped weight value W[n][t] = clip(w,-1,1)*clip(wf,1,255); 0 outside 4x9
__device__ __forceinline__ float wclamp(const float* __restrict__ w,
                                        const float* __restrict__ wf,
                                        int n, int t) {
  if (n >= 4 || t >= 9) return 0.0f;
  float ww = fminf(fmaxf(w[n * 9 + t], -1.0f), 1.0f);
  float ff = fminf(fmaxf(wf[n], 1.0f), 255.0f);
  return ww * ff;
}

// One 16-pixel tile (tile t: y = t/4, x0 = (t%4)*16) through 3 chained
// V_WMMA_F32_16X16X4_F32: A = kernels(16x4 taps), B = patches(4x16 pixels).
// Result: lanes 0-15, c[0..3] = kernel 0..3 output for pixel (lane&15).
__device__ __forceinline__ v8f conv_tile(const float* tile, int t,
                                         v2f a0, v2f a1, v2f a2) {
  const int lane = threadIdx.x & 31;
  const int pix  = lane & 15;
  const int half = lane >> 4;
  const int y    = t >> 2;
  const int x    = ((t & 3) << 4) + pix;
  const float* base = tile + y * PW + x + 3;   // tap(tt) = base[(tt/3)*PW + tt%3]
  // B VGPR0 holds K rows {0,2} (lanes 0-15 / 16-31), VGPR1 rows {1,3}
  int o00 = half ? 2          : 0;        // j=0: taps {0,1} | {2,3}
  int o01 = half ? PW         : 1;
  int o10 = half ? 2 * PW     : PW + 1;   // j=1: taps {4,5} | {6,7}
  int o11 = half ? 2 * PW + 1 : PW + 2;
  float t8 = base[2 * PW + 2];
  v2f b0, b1, b2;
  b0.x = base[o00]; b0.y = base[o01];
  b1.x = base[o10]; b1.y = base[o11];
  b2.x = half ? 0.0f : t8;  b2.y = 0.0f;  // j=2: taps {8,-} | {-,-}
  v8f c = {0.f, 0.f, 0.f, 0.f, 0.f, 0.f, 0.f, 0.f};
  c = __builtin_amdgcn_wmma_f32_16x16x4_f32(false, a0, false, b0, (short)0, c, false, false);
  c = __builtin_amdgcn_wmma_f32_16x16x4_f32(false, a1, false, b1, (short)0, c, false, false);
  c = __builtin_amdgcn_wmma_f32_16x16x4_f32(false, a2, false, b2, (short)0, c, false, false);
  return c;
}

__device__ __forceinline__ void load_A(const float* __restrict__ w,
                                       const float* __restrict__ wf,
                                       v2f& a0, v2f& a1, v2f& a2) {
  const int lane = threadIdx.x & 31;
  const int n = lane & 15, half = lane >> 4, k0 = half << 1;
  a0.x = wclamp(w, wf, n, 0 + k0); a0.y = wclamp(w, wf, n, 1 + k0);
  a1.x = wclamp(w, wf, n, 4 + k0); a1.y = wclamp(w, wf, n, 5 + k0);
  a2.x = half ? 0.0f : wclamp(w, wf, n, 8); a2.y = 0.0f;
}

__global__ void kinit(unsigned int* ws) {
  if (threadIdx.x < 4) ws[threadIdx.x] = 0u;   // key(-inf-ish) == smallest key
}

// Pass 1: per-kernel global max of conv result (uint-key atomicMax).
__global__ void __launch_bounds__(256) kmax(const float* __restrict__ inp,
                                            const float* __restrict__ w,
                                            const float* __restrict__ wf,
                                            unsigned int* __restrict__ ws) {
  __shared__ float tile[TILE_FLOATS];
  __shared__ float red[8][4];
  const int tid = threadIdx.x, lane = tid & 31, wave = tid >> 5;
  const int half = lane >> 4;
  const int ch = blockIdx.x;

  v2f a0, a1, a2;
  load_A(w, wf, a0, a1, a2);
  fill_tile_async(tile, inp + (size_t)ch * (IMG * IMG));

  float m0 = -3.4e38f, m1 = -3.4e38f, m2 = -3.4e38f, m3 = -3.4e38f;
#pragma unroll 4
  for (int k = 0; k < 32; ++k) {
    v8f c = conv_tile(tile, wave * 32 + k, a0, a1, a2);
    // lanes >=16 hold kernel rows 8..15 (zero weights) -> mask with -inf
    m0 = fmaxf(m0, half ? -3.4e38f : c[0]);
    m1 = fmaxf(m1, half ? -3.4e38f : c[1]);
    m2 = fmaxf(m2, half ? -3.4e38f : c[2]);
    m3 = fmaxf(m3, half ? -3.4e38f : c[3]);
  }
#pragma unroll
  for (int off = 16; off > 0; off >>= 1) {
    m0 = fmaxf(m0, __shfl_xor(m0, off, 32));
    m1 = fmaxf(m1, __shfl_xor(m1, off, 32));
    m2 = fmaxf(m2, __shfl_xor(m2, off, 32));
    m3 = fmaxf(m3, __shfl_xor(m3, off, 32));
  }
  if (lane == 0) { red[wave][0] = m0; red[wave][1] = m1; red[wave][2] = m2; red[wave][3] = m3; }
  __syncthreads();
  if (tid < 4) {
    float mm = red[0][tid];
#pragma unroll
    for (int wv = 1; wv < 8; ++wv) mm = fmaxf(mm, red[wv][tid]);
    unsigned b   = __float_as_uint(mm);
    unsigned key = (b & 0x80000000u) ? ~b : (b | 0x80000000u);  // monotone map
    atomicMax(&ws[tid], key);
  }
}

// Pass 2: recompute conv, apply power-of-two scale, add input, write out.
__global__ void __launch_bounds__(256) kapply(const float* __restrict__ inp,
                                              const float* __restrict__ w,
                                              const float* __restrict__ wf,
                                              const unsigned int* __restrict__ ws,
                                              float* __restrict__ out) {
  __shared__ float tile[TILE_FLOATS];
  const int tid = threadIdx.x, lane = tid & 31, wave = tid >> 5;
  const int ch = blockIdx.x;

  v2f a0, a1, a2;
  load_A(w, wf, a0, a1, a2);

  float scl[4];
#pragma unroll
  for (int i = 0; i < 4; ++i) {
    unsigned key = ws[i];
    unsigned b   = (key & 0x80000000u) ? (key & 0x7FFFFFFFu) : ~key;
    float mx  = __uint_as_float(b);
    float div = mx * (1.0f / 128.0f);
    scl[i] = (div > 0.0f) ? exp2f(-rintf(log2f(div))) : 1.0f;  // snap to 2^-round(log2)
  }

  fill_tile_async(tile, inp + (size_t)ch * (IMG * IMG));

#pragma unroll 2
  for (int k = 0; k < 16; ++k) {
    int t0 = (wave * 16 + k) * 2;                 // pair of adjacent 16-pixel tiles
    v8f c0 = conv_tile(tile, t0,     a0, a1, a2);
    v8f c1 = conv_tile(tile, t0 + 1, a0, a1, a2);
    int pg = t0 * 16 + lane;                      // 32 consecutive pixels
    int y = pg >> 6, x = pg & 63;
    float inv = tile[(y + 1) * PW + (x + 4)];     // input value from LDS
    size_t obase = ((size_t)ch << 12) + (size_t)pg;
#pragma unroll
    for (int nn = 0; nn < 4; ++nn) {
      float va = c0[nn];
      float vb = __shfl(c1[nn], lane & 15, 32);   // tile1 results into upper lanes
      float v  = (lane < 16) ? va : vb;
      out[((size_t)nn << 24) + obase] = inv + v * scl[nn];  // full-wave 128B store
    }
  }
}

extern "C" void kernel_launch(void* const* d_in, const int* in_sizes, int n_in,
                              void* d_out, int out_size, void* d_ws, size_t ws_size,
                              hipStream_t stream) {
  const float* inp = (const float*)d_in[0];   // [1,4096,64,64]
  const float* w   = (const float*)d_in[1];   // [4,9]
  const float* wf  = (const float*)d_in[2];   // [4,1]
  unsigned int* ws = (unsigned int*)d_ws;     // 4 max-keys
  float* out = (float*)d_out;                 // [4,1,4096,64,64]

  hipLaunchKernelGGL(kinit,  dim3(1),  dim3(64),  0, stream, ws);
  hipLaunchKernelGGL(kmax,   dim3(CH), dim3(256), 0, stream, inp, w, wf, ws);
  hipLaunchKernelGGL(kapply, dim3(CH), dim3(256), 0, stream, inp, w, wf, ws, out);
}